// GenericEncoderDecoder_43130061586700
// MI455X (gfx1250) — compile-verified
//
#include <hip/hip_runtime.h>
#include <hip/hip_bf16.h>
#include <stdint.h>

// Problem constants (match reference)
#define B_TOK  4096
#define KTOP   64
#define DIN    4096
#define DSAE   65536

// Tiling
#define NBUF   4          // async pipeline depth (LDS buffers)
#define CH     256        // contraction chunks per tile: 16 tokens * 16 k-chunks of 4
#define COLS_PER_WAVE 32  // two 16-wide N tiles per wave

typedef float v2f __attribute__((ext_vector_type(2)));
typedef float v8f __attribute__((ext_vector_type(8)));

// Single wave per workgroup: ASYNCcnt sequences the LDS pipeline, no barriers.
__global__ __launch_bounds__(32)
void sae_decode_wmma(const int*   __restrict__ idx,   // [B, K] feature indices
                     const float* __restrict__ val,   // [B, K] activations
                     const float* __restrict__ W,     // [DSAE, DIN]
                     const float* __restrict__ bdec,  // [DIN]
                     float*       __restrict__ out)   // [B, DIN]
{
    // 4 buffers * (4 rows * 32 cols) floats = 2 KB static LDS
    __shared__ __align__(16) float smem[NBUF * 4 * COLS_PER_WAVE];

    const int lane    = threadIdx.x;        // 0..31
    const int colBase = blockIdx.x * COLS_PER_WAVE;
    const int tokBase = blockIdx.y * 16;

    const int*   idxT = idx + tokBase * KTOP;
    const float* valT = val + tokBase * KTOP;

    // ---- copy-phase lane mapping: each lane moves 16B of a gathered row ----
    const int crow = lane >> 3;              // which of the 4 gathered rows (0..3)
    const int coff = (lane & 7) * 4;         // float offset inside 32-col segment

    // LDS byte address of this lane's 16B slot in buffer 0.
    // Flat shared address truncated to 32 bits == LDS byte offset.
    // Taking smem's address here also forces the array to escape, so the ds reads stay real.
    const uint32_t ldsBase0 = (uint32_t)(uintptr_t)(&smem[0]) + (uint32_t)lane * 16u;

    // ---- compute-phase lane mapping (ISA 16x16 f32 fragment striping) ----
    const int half = lane >> 4;              // 0: K rows {0,1} / M 0..7 ; 1: K rows {2,3} / M 8..15
    const int mcol = lane & 15;              // M for A, N for B/C

    v8f acc0 = {};                           // C tile, cols [0,16)
    v8f acc1 = {};                           // C tile, cols [16,32)

    auto issue = [&](int c, int buf) {
        const int t  = c >> 4;
        const int kc = (c & 15) << 2;
        // Wave-uniform load of the 4 gathered-row indices (s_load_b128), then
        // per-lane select of this lane's row index (v_cndmask, no VMEM gather).
        const int4 gi4 = *(const int4*)(idxT + t * KTOP + kc);
        const int giLo = (crow & 1) ? gi4.y : gi4.x;
        const int giHi = (crow & 1) ? gi4.w : gi4.z;
        const int gi   = (crow & 2) ? giHi : giLo;

        const unsigned long long src =
            (unsigned long long)(uintptr_t)(W + (size_t)(uint32_t)gi * DIN + colBase + coff);
        const uint32_t dst = ldsBase0 + (uint32_t)(buf * (4 * COLS_PER_WAVE) * 4);
        // CDNA5 async copy: global -> LDS, tracked on ASYNCcnt
        asm volatile("global_load_async_to_lds_b128 %0, %1, off"
                     :: "v"(dst), "v"(src) : "memory");
    };

    // Prologue: fill the pipeline
    #pragma unroll
    for (int p = 0; p < NBUF; ++p) issue(p, p);

    for (int c = 0; c < CH; ++c) {
        const int buf = c & (NBUF - 1);

        // Wait until chunk c's copy (the oldest in flight) has landed in LDS.
        int w = CH - 1 - c; if (w > NBUF - 1) w = NBUF - 1;
        switch (w) {
            case 3:  asm volatile("s_wait_asynccnt 0x3" ::: "memory"); break;
            case 2:  asm volatile("s_wait_asynccnt 0x2" ::: "memory"); break;
            case 1:  asm volatile("s_wait_asynccnt 0x1" ::: "memory"); break;
            default: asm volatile("s_wait_asynccnt 0x0" ::: "memory"); break;
        }

        const int t  = c >> 4;
        const int kc = (c & 15) << 2;
        const int bb = buf * (4 * COLS_PER_WAVE);

        // B fragments (4x16 f32) from LDS: VGPR0 holds K-rows {0 | 2}, VGPR1 K-rows {1 | 3}
        const int r0 = half * 2;
        v2f B0, B1;
        B0.x = smem[bb + (r0    ) * COLS_PER_WAVE + mcol];
        B0.y = smem[bb + (r0 + 1) * COLS_PER_WAVE + mcol];
        B1.x = smem[bb + (r0    ) * COLS_PER_WAVE + 16 + mcol];
        B1.y = smem[bb + (r0 + 1) * COLS_PER_WAVE + 16 + mcol];

        // A fragment (16x4 f32), block-diagonal: only row t carries values.
        // Uniform s_load of the 4 activations, then VALU selects (no predicated VMEM).
        const float4 v4 = *(const float4*)(valT + t * KTOP + kc);
        float a0 = half ? v4.z : v4.x;       // K = 0|2
        float a1 = half ? v4.w : v4.y;       // K = 1|3
        a0 = (mcol == t) ? a0 : 0.0f;
        a1 = (mcol == t) ? a1 : 0.0f;
        v2f A; A.x = a0; A.y = a1;

        acc0 = __builtin_amdgcn_wmma_f32_16x16x4_f32(false, A, false, B0,
                                                     (short)0, acc0, false, false);
        acc1 = __builtin_amdgcn_wmma_f32_16x16x4_f32(false, A, false, B1,
                                                     (short)0, acc1, false, false);

        // Refill the buffer we just consumed (ds reads have completed into VGPRs
        // before this asm: compiler inserts the dscnt wait ahead of the WMMAs).
        if (c + NBUF < CH) issue(c + NBUF, buf);
    }

    // Epilogue: add b_dec and store the 16x32 tile (coalesced 64B per half-wave row)
    const float bd0 = bdec[colBase + mcol];
    const float bd1 = bdec[colBase + 16 + mcol];
    float* op = out + (size_t)tokBase * DIN + colBase;
    #pragma unroll
    for (int r = 0; r < 8; ++r) {
        const int row = r + half * 8;        // C layout: VGPR r -> M = r (+8 for lanes 16..31)
        op[(size_t)row * DIN + mcol]      = acc0[r] + bd0;
        op[(size_t)row * DIN + 16 + mcol] = acc1[r] + bd1;
    }
}

extern "C" void kernel_launch(void* const* d_in, const int* in_sizes, int n_in,
                              void* d_out, int out_size, void* d_ws, size_t ws_size,
                              hipStream_t stream) {
    const int*   idx  = (const int*)  d_in[0];   // indices [B, K]
    const float* val  = (const float*)d_in[1];   // values  [B, K]
    const float* W    = (const float*)d_in[2];   // W_dec   [DSAE, DIN]
    const float* bdec = (const float*)d_in[3];   // b_dec   [DIN]
    float* out = (float*)d_out;                  // [B, DIN]

    dim3 grid(DIN / COLS_PER_WAVE, B_TOK / 16);  // 128 x 256 single-wave workgroups
    sae_decode_wmma<<<grid, 32, 0, stream>>>(idx, val, W, bdec, out);
}